// Model_44160853738111
// MI455X (gfx1250) — compile-verified
//
#include <hip/hip_runtime.h>
#include <math.h>

#define NSAMP   131072
#define NEV     128
#define NATOMS  32
#define ASIZE   512
#define LATENT  16
#define LAYERS  7
#define PI_D    3.14159265358979323846
// padded length N3 = 3*NSAMP = 393216, K = N3/2 = 196608, NC = K+1 = 196609

typedef float v2f __attribute__((ext_vector_type(2)));
typedef float v8f __attribute__((ext_vector_type(8)));
typedef unsigned int u32x4 __attribute__((ext_vector_type(4)));
typedef int i32x4 __attribute__((ext_vector_type(4)));
typedef int i32x8 __attribute__((ext_vector_type(8)));

// ---------------------------------------------------------------------------
// Tensor Data Mover: DMA `nelem` f32 (one row) from global to LDS.
// D# built per CDNA5 ISA 8.3/8.4: count=1, type=2, data_size=4B,
// tile = nelem x 1, tensor_dim0 = stride = nelem. Issued per-wave (EXEC
// ignored); completion via TENSORcnt.
// ---------------------------------------------------------------------------
__device__ __forceinline__ void tdm_load_f32(void* lds_dst, const void* gsrc,
                                             unsigned nelem)
{
    const unsigned long long ga = (unsigned long long)(uintptr_t)gsrc;
    const unsigned           la = (unsigned)(uintptr_t)lds_dst;
    u32x4 g0;
    g0.x = 1u;                                                  // count=1
    g0.y = la;                                                  // lds_addr
    g0.z = (unsigned)(ga & 0xFFFFFFFFu);                        // gaddr[31:0]
    g0.w = (unsigned)((ga >> 32) & 0x01FFFFFFu) | (2u << 30);   // gaddr[56:32]|type=2
    i32x8 g1;
    g1[0] = (int)(2u << 16);                       // wg_mask=0, data_size=4B
    g1[1] = (int)((nelem & 0xFFFFu) << 16);        // tensor_dim0[15:0]
    g1[2] = (int)((nelem >> 16) | (1u << 16));     // tensor_dim0[31:16]|tensor_dim1=1
    g1[3] = (int)((nelem & 0xFFFFu) << 16);        // tile_dim0 = nelem
    g1[4] = 1;                                     // tile_dim1 = 1
    g1[5] = (int)nelem;                            // tensor_dim0_stride[31:0]
    g1[6] = 0;
    g1[7] = 0;
    i32x4 z4 = {0, 0, 0, 0};
#if __has_include(<hip/amd_detail/amd_gfx1250_TDM.h>)
    i32x8 z8 = {0, 0, 0, 0, 0, 0, 0, 0};
    __builtin_amdgcn_tensor_load_to_lds(g0, g1, z4, z4, z8, 0);
#else
    __builtin_amdgcn_tensor_load_to_lds(g0, g1, z4, z4, 0);
#endif
}

// ---------------------------------------------------------------------------
// Kernel 1: 7-layer latent expansion, per-event shift constants, amps, mix.
// Single block of 128 threads; everything lives in LDS.
// ---------------------------------------------------------------------------
__global__ void k_setup(const float* __restrict__ base_latent, // [16]
                        const float* __restrict__ to_off_w,    // [7][2][16]
                        const float* __restrict__ split_w,     // [7][32][16]
                        const float* __restrict__ split_b,     // [7][32]
                        const float* __restrict__ to_atoms_w,  // [32][16]
                        const float* __restrict__ to_atoms_b,  // [32]
                        const float* __restrict__ to_amp_w,    // [16]
                        const float* __restrict__ to_amp_b,    // [1]
                        float* __restrict__ mix,               // [128][32]
                        float* __restrict__ consts,            // [128][8]
                        float* __restrict__ amp)               // [128]
{
    __shared__ float xbuf[2][NEV][LATENT];
    __shared__ float tbuf[2][NEV];
    const int tid = threadIdx.x;   // blockDim.x == 128

    if (tid < LATENT) xbuf[0][0][tid] = base_latent[tid];
    if (tid == 0)     tbuf[0][0] = 0.0f;
    __syncthreads();

    int cur = 0;
    for (int i = 0; i < LAYERS; ++i) {
        const int   n     = 1 << i;
        const float scale = 1.0f / (float)(1 << i);
        const int   nxt   = cur ^ 1;
        for (int en = tid; en < 2 * n; en += blockDim.x) {
            const int e = en >> 1, o = en & 1;
            const float* w = to_off_w + (i * 2 + o) * LATENT;
            float acc = 0.0f;
            for (int l = 0; l < LATENT; ++l) acc += xbuf[cur][e][l] * w[l];
            const float sg = 1.0f / (1.0f + expf(-acc));
            tbuf[nxt][en] = tbuf[cur][e] + sg * scale;
        }
        for (int idx = tid; idx < 2 * n * LATENT; idx += blockDim.x) {
            const int en = idx >> 4, j = idx & 15;
            const int e = en >> 1, o = en & 1;
            const float* w = split_w + ((i * 32) + o * 16 + j) * LATENT;
            float acc = split_b[i * 32 + o * 16 + j];
            for (int l = 0; l < LATENT; ++l) acc += xbuf[cur][e][l] * w[l];
            xbuf[nxt][en][j] = acc;
        }
        __syncthreads();
        cur = nxt;
    }

    // per-event amp + fractional-shift phase constants (tid == event)
    {
        const int e = tid;
        float a = to_amp_b[0];
        for (int l = 0; l < LATENT; ++l) a += xbuf[cur][e][l] * to_amp_w[l];
        amp[e] = a;

        const double s   = (double)tbuf[cur][e] * 65536.0;   // shift_samples
        const double phi = PI_D * s / 196609.0;              // pi*s/NC
        const double th1 = fmod((2.0 * 196608.0 - 1.0) * phi, 2.0 * PI_D);
        const double thN = fmod((2.0 * 196608.0)       * phi, 2.0 * PI_D);
        consts[e * 8 + 0] = (float)cos(phi);
        consts[e * 8 + 1] = (float)sin(phi);
        consts[e * 8 + 2] = (float)cos(th1);
        consts[e * 8 + 3] = (float)sin(th1);
        consts[e * 8 + 4] = (float)cos(thN);
        consts[e * 8 + 5] = 0.0f;
        consts[e * 8 + 6] = 0.0f;
        consts[e * 8 + 7] = 0.0f;
    }

    for (int idx = tid; idx < NEV * NATOMS; idx += blockDim.x) {
        const int e = idx >> 5, a = idx & 31;
        float acc = to_atoms_b[a];
        for (int l = 0; l < LATENT; ++l) acc += xbuf[cur][e][l] * to_atoms_w[a * LATENT + l];
        mix[e * NATOMS + a] = acc;
    }
}

// ---------------------------------------------------------------------------
// Kernel 2: ev[e][t] = (mix · atoms)[e][t] * hamming(t), via fp32 WMMA.
// One wave per 16x16 tile, K=32 in 8 steps of V_WMMA_F32_16X16X4_F32.
// ---------------------------------------------------------------------------
__global__ void k_atoms_wmma(const float* __restrict__ mix,    // [128][32]
                             const float* __restrict__ atoms,  // [32][512]
                             float* __restrict__ ev)           // [128][512]
{
    const int lane = threadIdx.x;       // 0..31
    const int half = lane >> 4;
    const int lr   = lane & 15;
    const int t0   = blockIdx.x * 16;
    const int e0   = blockIdx.y * 16;

    v8f c = {};
    for (int k0 = 0; k0 < NATOMS; k0 += 4) {
        v2f a, b;
        a.x = mix[(e0 + lr) * NATOMS + k0 + 2 * half + 0];
        a.y = mix[(e0 + lr) * NATOMS + k0 + 2 * half + 1];
        b.x = atoms[(k0 + 2 * half + 0) * ASIZE + t0 + lr];
        b.y = atoms[(k0 + 2 * half + 1) * ASIZE + t0 + lr];
        c = __builtin_amdgcn_wmma_f32_16x16x4_f32(false, a, false, b,
                                                  (short)0, c, false, false);
    }

    const float tf = (float)(t0 + lr);
    const float w  = 0.54f - 0.46f * __cosf(6.283185307179586f * tf / 512.0f);
    for (int r = 0; r < 8; ++r) {
        const int e = e0 + r + 8 * half;
        ev[e * ASIZE + t0 + lr] = c[r] * w;
    }
}

// ---------------------------------------------------------------------------
// Kernel 3: per-event L2 normalize (+1e-8) and amplitude scale, in place.
// ---------------------------------------------------------------------------
__global__ void k_norm(const float* __restrict__ amp, float* __restrict__ ev)
{
    __shared__ float red[256];
    const int e = blockIdx.x, tid = threadIdx.x;
    const float a0 = ev[e * ASIZE + tid];
    const float a1 = ev[e * ASIZE + 256 + tid];
    red[tid] = a0 * a0 + a1 * a1;
    __syncthreads();
    for (int s = 128; s > 0; s >>= 1) {
        if (tid < s) red[tid] += red[tid + s];
        __syncthreads();
    }
    const float scale = amp[e] / (sqrtf(red[0]) + 1e-8f);
    ev[e * ASIZE + tid]       = a0 * scale;
    ev[e * ASIZE + 256 + tid] = a1 * scale;
}

// ---------------------------------------------------------------------------
// Kernel 4: fractional-shift synthesis via the exact Dirichlet closed form of
// pad(3N)->rfft->phase->irfft->crop.  out[m] = (-1)^m/N3 *
//   Σ_e Σ_t v_e[t]·(-1)^t·[ (s1·cosβ - c1·sinβ)/(c0·sinβ + s0·cosβ) + cN ]
// β = π(m-t)/N3 by incremental rotation.  v rows double-buffered into LDS by
// the Tensor Data Mover (wave 0 issues, TENSORcnt + barrier publishes).
// ---------------------------------------------------------------------------
__global__ void __launch_bounds__(256) k_synth(
                        const float* __restrict__ v,       // [128][512]
                        const float* __restrict__ consts,  // [128][8]
                        float* __restrict__ out)           // [131072]
{
    __shared__ float lcons[NEV * 8];    // 4KB: per-event phase constants
    __shared__ float vbuf[2][ASIZE];    // 2x2KB: double-buffered event rows
    const int  tid   = threadIdx.x;
    const int  m     = blockIdx.x * blockDim.x + tid;
    const bool wave0 = (tid < 32);

    if (wave0) {
        tdm_load_f32(lcons, consts, NEV * 8);
        tdm_load_f32(vbuf[0], v, ASIZE);
        __builtin_amdgcn_s_wait_tensorcnt(0);
    }
    __syncthreads();

    const float dlt  = (float)(PI_D / 393216.0);       // β step per t
    const float sdlt = (float)sin(PI_D / 393216.0);
    const float cdlt = (float)cos(PI_D / 393216.0);    // == 1.0f in fp32
    const float b0   = (float)m * dlt;                 // β at t = 0

    float acc = 0.0f;
    for (int e = 0; e < NEV; ++e) {
        // prefetch next event row behind this event's compute
        if (wave0 && (e + 1 < NEV))
            tdm_load_f32(vbuf[(e + 1) & 1], v + (e + 1) * ASIZE, ASIZE);

        const float c0 = lcons[e * 8 + 0];
        const float s0 = lcons[e * 8 + 1];
        const float c1 = lcons[e * 8 + 2];
        const float s1 = lcons[e * 8 + 3];
        const float cN = lcons[e * 8 + 4];
        float sb, cb;
        __sincosf(b0, &sb, &cb);
        const float* vp = vbuf[e & 1];
#pragma unroll 4
        for (int t = 0; t < ASIZE; t += 2) {
            // even t  (sign +)
            float num = fmaf(cb, s1, -sb * c1);
            float den = fmaf(sb, c0,  cb * s0);
            float g   = fmaf(num, __builtin_amdgcn_rcpf(den), cN);
            acc = fmaf(vp[t], g, acc);
            float cb2 = fmaf(cb, cdlt,  sb * sdlt);    // rotate β -= δ
            float sb2 = fmaf(sb, cdlt, -cb * sdlt);
            // odd t  (sign -)
            float num2 = fmaf(cb2, s1, -sb2 * c1);
            float den2 = fmaf(sb2, c0,  cb2 * s0);
            float g2   = fmaf(num2, __builtin_amdgcn_rcpf(den2), cN);
            acc = fmaf(-vp[t + 1], g2, acc);
            cb = fmaf(cb2, cdlt,  sb2 * sdlt);
            sb = fmaf(sb2, cdlt, -cb2 * sdlt);
        }

        if (wave0) __builtin_amdgcn_s_wait_tensorcnt(0);
        __syncthreads();   // publish next buffer; gate reuse of current one
    }
    const float sgn = (m & 1) ? -1.0f : 1.0f;
    out[m] = acc * sgn * (1.0f / 393216.0f);
}

// ---------------------------------------------------------------------------
extern "C" void kernel_launch(void* const* d_in, const int* in_sizes, int n_in,
                              void* d_out, int out_size, void* d_ws, size_t ws_size,
                              hipStream_t stream)
{
    const float* base_latent = (const float*)d_in[0];  // 16
    const float* to_off_w    = (const float*)d_in[1];  // 7*2*16
    const float* split_w     = (const float*)d_in[2];  // 7*32*16
    const float* split_b     = (const float*)d_in[3];  // 7*32
    const float* atoms       = (const float*)d_in[4];  // 32*512
    const float* to_atoms_w  = (const float*)d_in[5];  // 32*16
    const float* to_atoms_b  = (const float*)d_in[6];  // 32
    const float* to_amp_w    = (const float*)d_in[7];  // 16
    const float* to_amp_b    = (const float*)d_in[8];  // 1
    float* out = (float*)d_out;                        // 131072

    // workspace layout (floats): mix 4096 | ev 65536 | consts 1024 | amp 128
    float* ws_mix    = (float*)d_ws;
    float* ws_ev     = ws_mix + NEV * NATOMS;
    float* ws_consts = ws_ev + NEV * ASIZE;
    float* ws_amp    = ws_consts + NEV * 8;

    k_setup<<<1, 128, 0, stream>>>(base_latent, to_off_w, split_w, split_b,
                                   to_atoms_w, to_atoms_b, to_amp_w, to_amp_b,
                                   ws_mix, ws_consts, ws_amp);

    k_atoms_wmma<<<dim3(ASIZE / 16, NEV / 16), 32, 0, stream>>>(ws_mix, atoms, ws_ev);

    k_norm<<<NEV, 256, 0, stream>>>(ws_amp, ws_ev);

    k_synth<<<NSAMP / 256, 256, 0, stream>>>(ws_ev, ws_consts, out);
}